// M1Net_25288767438964
// MI455X (gfx1250) — compile-verified
//
#include <hip/hip_runtime.h>
#include <hip/hip_bf16.h>

// ---------------------------------------------------------------------------
// Izhikevich SNN firing-rate, MI455X-optimized.
//
// I = round(x*16) takes only 17 integer values and all neurons share initial
// state, so the 200-step simulation is a pure function of I: build a 17-entry
// firing-rate LUT (kernel 1, ~54 kflops), then stream 64 MB in + 64 MB out
// (kernel 2, HBM-bound, ~5.5 us @ 23.3 TB/s).
//
// gfx1250 paths used: global_load_async_to_lds_b128 (ASYNCcnt) double-buffered
// pipeline, s_wait_asynccnt fences, ds_bpermute_b32 crossbar LUT lookup,
// last-use/non-temporal cache hints on the single-pass stream.
// ---------------------------------------------------------------------------

#define NSTEPS 200
#define BLK    256
#define NBLOCKS 2048

// --- Kernel 1: build the 17-entry LUT (exact reference update ordering) -----
__global__ __launch_bounds__(32) void izh_build_lut(float* __restrict__ lut) {
  const int i = (int)threadIdx.x;
  const float I = (float)(i <= 16 ? i : 16);
  float mem = -70.0f;   // param_mem
  float u   = -14.0f;   // param_u
  float acc = 0.0f;
  for (int t = 0; t < NSTEPS; ++t) {
    // mem += dt*(0.04 mem^2 + 5 mem + 140 - u + I)
    mem = mem + 0.5f * (0.04f * mem * mem + 5.0f * mem + 140.0f - u + I);
    // u += dt*a*(b*mem_new - u)   (uses updated mem, before reset)
    u = u + (0.5f * 0.02f) * (0.2f * mem - u);
    const bool s = mem > 30.0f;
    acc = s ? acc + 1.0f : acc;
    mem = s ? -65.0f : mem;      // reset c
    u   = s ? u + 8.0f : u;      // u += d
  }
  if (i <= 16) lut[i] = acc * (1.0f / (float)NSTEPS);
}

// --- gfx1250 async global->LDS load path ------------------------------------
typedef int   v4i __attribute__((__vector_size__(16)));
typedef float v4f __attribute__((ext_vector_type(4)));
typedef __attribute__((address_space(1))) v4i as1_v4i;   // global
typedef __attribute__((address_space(3))) v4i as3_v4i;   // LDS

#if __has_builtin(__builtin_amdgcn_global_load_async_to_lds_b128)
#define USE_ASYNC_BUILTIN 1
#else
#define USE_ASYNC_BUILTIN 0
#endif

// cpol TH[2:0]=3 -> LU (last-use: non-temporal, discard on hit). Single-pass input.
__device__ __forceinline__ void issue_async16(const float4* g, float4* l) {
#if USE_ASYNC_BUILTIN
  __builtin_amdgcn_global_load_async_to_lds_b128(
      (as1_v4i*)(void*)g, (as3_v4i*)(void*)l, /*offset=*/0, /*cpol=*/3);
#else
  unsigned lofs = (unsigned)(__SIZE_TYPE__)l;
  asm volatile("global_load_async_to_lds_b128 %0, %1, off th:TH_LOAD_LU"
               :: "v"(lofs), "v"((unsigned long long)(const void*)g)
               : "memory");
#endif
}

#define WAIT_ASYNC_LE_1() asm volatile("s_wait_asynccnt 0x1" ::: "memory")
#define WAIT_ASYNC_LE_0() asm volatile("s_wait_asynccnt 0x0" ::: "memory")

// 17-way LUT lookup via register crossbar: lane k (k<=16) holds lut[k].
__device__ __forceinline__ float lut_lookup(float v, int lutBits) {
  const int idx = (int)rintf(v * 16.0f);   // v_rndne_f32: round-half-even, matches jnp.round
  return __int_as_float(__builtin_amdgcn_ds_bpermute(idx << 2, lutBits));
}

// --- Kernel 2: streaming map -------------------------------------------------
__global__ __launch_bounds__(BLK) void izh_map(const float* __restrict__ x,
                                               float* __restrict__ out,
                                               const float* __restrict__ lut,
                                               int n) {
  __shared__ float4 sbuf[2][BLK];

  const unsigned t     = threadIdx.x;
  const unsigned gtid  = blockIdx.x * BLK + t;
  const unsigned total = gridDim.x * BLK;
  const int lane = (int)(t & 31u);
  // Every lane loads its LUT slot once (L2-hot); lanes 17..31 hold a dummy.
  const int lutBits = __float_as_int(lut[lane <= 16 ? lane : 16]);

  const int n4     = n >> 2;
  const int chunks = n4 / (int)total;          // uniform full-tile chunks
  const float4* __restrict__ x4 = (const float4*)x;
  v4f* __restrict__ o4          = (v4f*)out;

  // Double-buffered async global->LDS pipeline (512 B per wave per issue).
  if (chunks > 0) issue_async16(x4 + (size_t)0 * total + gtid, &sbuf[0][t]);
  if (chunks > 1) issue_async16(x4 + (size_t)1 * total + gtid, &sbuf[1][t]);

  for (int k = 0; k < chunks; ++k) {
    if (k + 1 < chunks) { WAIT_ASYNC_LE_1(); } else { WAIT_ASYNC_LE_0(); }
    const float4 v = sbuf[k & 1][t];           // ds_load_b128 from own slot
    v4f r;
    r.x = lut_lookup(v.x, lutBits);
    r.y = lut_lookup(v.y, lutBits);
    r.z = lut_lookup(v.z, lutBits);
    r.w = lut_lookup(v.w, lutBits);
    if (k + 2 < chunks)                        // refill the slot just consumed
      issue_async16(x4 + (size_t)(k + 2) * total + gtid, &sbuf[k & 1][t]);
    // Output is write-once: non-temporal store (TH=NT), no L2 write-allocate churn.
    __builtin_nontemporal_store(r, o4 + (size_t)k * total + gtid);
  }

  // float4 tail (reconvergent predication: ds_bpermute needs full EXEC).
  {
    const int i4  = chunks * (int)total + (int)gtid;
    const bool act = i4 < n4;
    v4f v = (v4f)(0.0f);
    if (act) v = __builtin_nontemporal_load((const v4f*)x4 + i4);
    v4f r;
    r.x = lut_lookup(v.x, lutBits);
    r.y = lut_lookup(v.y, lutBits);
    r.z = lut_lookup(v.z, lutBits);
    r.w = lut_lookup(v.w, lutBits);
    if (act) __builtin_nontemporal_store(r, o4 + i4);
  }
  // scalar tail (n not divisible by 4)
  {
    const int base = n4 << 2;
    const int i    = base + (int)gtid;
    const bool act = i < n;
    float v = 0.f;
    if (act) v = x[i];
    const float r = lut_lookup(v, lutBits);
    if (act) out[i] = r;
  }
}

// --- Launch ------------------------------------------------------------------
extern "C" void kernel_launch(void* const* d_in, const int* in_sizes, int n_in,
                              void* d_out, int out_size, void* d_ws, size_t ws_size,
                              hipStream_t stream) {
  const float* x = (const float*)d_in[0];
  float* out     = (float*)d_out;
  float* lut     = (float*)d_ws;          // 17 floats of scratch
  const int n    = in_sizes[0];

  izh_build_lut<<<1, 32, 0, stream>>>(lut);
  izh_map<<<NBLOCKS, BLK, 0, stream>>>(x, out, lut, n);
}